// AutoCorrelation_72138270704104
// MI455X (gfx1250) — compile-verified
//
#include <hip/hip_runtime.h>
#include <math.h>

// ---------------------------------------------------------------------------
// AutoCorrelation + GCN forward for MI455X (gfx1250, wave32).
// Shapes: B=4, N=325, T=OUT_LEN=288, F=64, H=FPH=8, PATCH=12, L=24, TOPK=5.
//
// Algebraic collapse (FPH==H==8): v[b,n,g,h,t] == v_small[b,n,t,h], so
//   out[b,n,t,f] = s[b,n,t] * wsum[f],   wsum[f] = sum_c Wout_map[f,c]
//   s[b,n,t]     = (1/8) sum_h sum_i tmp_corr[h,i] * v_small[b,n,(delay+t)%T, h]
// Dominant dense op: per (b,n),
//   new_values[t,f] = sum_{h<128} gelu(gw[h]*c[t]) * gWout[h,f]
// -> f32 WMMA (V_WMMA_F32_16X16X4_F32), G precomputed per 32-row chunk in LDS
// so the branchy exact-erf gelu never sits between WMMAs and is evaluated 1x.
// Problem is HBM-bound (~400 MB @ 23.3 TB/s ~ 17 us floor), so full f32.
// ---------------------------------------------------------------------------

#define B_     4
#define N_     325
#define T_     288
#define F_C    64
#define H_     8
#define PATCH_ 12
#define L_     24
#define TOPK_  5
#define FPH_   8
#define BN_    (B_ * N_)
#define ROWS_  (BN_ * T_)

typedef float v2f __attribute__((ext_vector_type(2)));
typedef float v8f __attribute__((ext_vector_type(8)));

__device__ __forceinline__ float gelu_exact(float x) {
  // approximate=False gelu: 0.5*x*(1+erf(x/sqrt(2)))
  return 0.5f * x * (1.0f + erff(x * 0.70710678118654752440f));
}

// ---------------------------------------------------------------------------
// K1: v_small[row, g] = sum_f V_in[row, f] * Wv[g, f]   (row = b,n,t flat)
// ---------------------------------------------------------------------------
__global__ __launch_bounds__(256) void k_vsmall(
    const float* __restrict__ V_in, const float* __restrict__ Wv,
    float* __restrict__ vs) {
  __shared__ float wv[FPH_ * F_C];
  for (int i = threadIdx.x; i < FPH_ * F_C; i += blockDim.x) wv[i] = Wv[i];
  __syncthreads();
  int row = blockIdx.x * blockDim.x + threadIdx.x;
  if (row >= ROWS_) return;
  const float4* vr4 = reinterpret_cast<const float4*>(V_in + (size_t)row * F_C);
  float acc[FPH_] = {};
#pragma unroll 4
  for (int f4 = 0; f4 < F_C / 4; ++f4) {
    float4 x = vr4[f4];
#pragma unroll
    for (int g = 0; g < FPH_; ++g) {
      const float* w = &wv[g * F_C + f4 * 4];
      acc[g] += x.x * w[0] + x.y * w[1] + x.z * w[2] + x.w * w[3];
    }
  }
  float4* o4 = reinterpret_cast<float4*>(vs + (size_t)row * FPH_);
  o4[0] = make_float4(acc[0], acc[1], acc[2], acc[3]);
  o4[1] = make_float4(acc[4], acc[5], acc[6], acc[7]);
}

// ---------------------------------------------------------------------------
// K2: per (b,n): q/k conv projections -> circular corr over L=24 -> top-5 ->
// softmax -> delay gather over v_small -> s[t].  288 threads.
// ---------------------------------------------------------------------------
__global__ __launch_bounds__(288) void k_corr(
    const float* __restrict__ Q_in, const float* __restrict__ K_in,
    const float* __restrict__ Wq, const float* __restrict__ Wk,
    const float* __restrict__ vs, float* __restrict__ s_out,
    float* __restrict__ delay_out, float* __restrict__ tc_out) {
  const int bn = blockIdx.x;
  const int tid = threadIdx.x;

  __shared__ float wq_s[H_ * F_C * PATCH_];   // 6144
  __shared__ float wk_s[H_ * F_C * PATCH_];   // 6144
  __shared__ float qm[H_][L_], km[H_][L_], corr[H_][L_];
  __shared__ float tc[H_][TOPK_];
  __shared__ int   dly[H_][TOPK_];
  __shared__ float vsl[T_ * FPH_];            // 2304

  for (int i = tid; i < H_ * F_C * PATCH_; i += blockDim.x) {
    wq_s[i] = Wq[i];
    wk_s[i] = Wk[i];
  }
  for (int i = tid; i < T_ * FPH_; i += blockDim.x)
    vsl[i] = vs[(size_t)bn * T_ * FPH_ + i];
  __syncthreads();

  // Stage A: q[h,l] = sum_{p,f} Q[l*12+p, f] * Wq[h, f, p]
  if (tid < H_ * L_) {
    const int h = tid / L_, l = tid % L_;
    const float* qrow = Q_in + ((size_t)bn * T_ + l * PATCH_) * F_C;
    const float* krow = K_in + ((size_t)bn * T_ + l * PATCH_) * F_C;
    const float* wqh = wq_s + h * F_C * PATCH_;
    const float* wkh = wk_s + h * F_C * PATCH_;
    float aq = 0.f, ak = 0.f;
    for (int p = 0; p < PATCH_; ++p)
      for (int f = 0; f < F_C; ++f) {
        aq += qrow[p * F_C + f] * wqh[f * PATCH_ + p];
        ak += krow[p * F_C + f] * wkh[f * PATCH_ + p];
      }
    qm[h][l] = aq;
    km[h][l] = ak;
  }
  __syncthreads();

  // Stage B: corr[h,d] = sum_l q[(l+d)%L] * k[l]
  if (tid < H_ * L_) {
    const int h = tid / L_, d = tid % L_;
    float c = 0.f;
    for (int l = 0; l < L_; ++l) {
      int li = l + d; if (li >= L_) li -= L_;
      c += qm[h][li] * km[h][l];
    }
    corr[h][d] = c;
  }
  __syncthreads();

  // Stage C: top-5 + softmax per head (lax.top_k tie-break: first index)
  if (tid < H_) {
    const int h = tid;
    float vals[L_];
    for (int i = 0; i < L_; ++i) vals[i] = corr[h][i];
    float w[TOPK_]; int idx[TOPK_];
    for (int i = 0; i < TOPK_; ++i) {
      int best = 0; float bv = -1e30f;
      for (int j = 0; j < L_; ++j)
        if (vals[j] > bv) { bv = vals[j]; best = j; }
      w[i] = bv; idx[i] = best; vals[best] = -1e30f;
    }
    float mx = w[0], sum = 0.f, e[TOPK_];
    for (int i = 0; i < TOPK_; ++i) { e[i] = expf(w[i] - mx); sum += e[i]; }
    const float inv = 1.0f / sum;
    for (int i = 0; i < TOPK_; ++i) {
      tc[h][i] = e[i] * inv;
      dly[h][i] = idx[i] * PATCH_;
      const size_t o = (size_t)bn * H_ * TOPK_ + h * TOPK_ + i;
      delay_out[o] = (float)(idx[i] * PATCH_);
      tc_out[o] = e[i] * inv;
    }
  }
  __syncthreads();

  // Stage D: s[t] = (1/H) sum_h sum_i tc[h,i] * v_small[(delay+t)%T, h]
  for (int t = tid; t < T_; t += blockDim.x) {
    float acc = 0.f;
#pragma unroll
    for (int h = 0; h < H_; ++h)
#pragma unroll
      for (int i = 0; i < TOPK_; ++i) {
        int g = dly[h][i] + t; if (g >= T_) g -= T_;
        acc += tc[h][i] * vsl[g * FPH_ + h];
      }
    s_out[(size_t)bn * T_ + t] = acc * (1.0f / H_);
  }
}

// ---------------------------------------------------------------------------
// K3: per (b,n): a_q/a_k reductions, Qg/Kg (128), Qa = Qg @ gWa.  128 threads.
// ---------------------------------------------------------------------------
__global__ __launch_bounds__(128) void k_gattn(
    const float* __restrict__ s, const float* __restrict__ Wout_map,
    const float* __restrict__ gQ1, const float* __restrict__ gQ2,
    const float* __restrict__ gK1, const float* __restrict__ gK2,
    const float* __restrict__ gWa, float* __restrict__ Qa_out,
    float* __restrict__ Kg_out) {
  const int bn = blockIdx.x, tid = threadIdx.x;
  __shared__ float sl[T_];
  __shared__ float red[128];
  __shared__ float wsum[F_C];
  __shared__ float qg[F_C], kg[F_C], Qg[128];

  for (int i = tid; i < T_; i += 128) sl[i] = s[(size_t)bn * T_ + i];
  if (tid < F_C) {
    float w = 0.f;
#pragma unroll
    for (int c = 0; c < FPH_; ++c) w += Wout_map[tid * FPH_ + c];
    wsum[tid] = w;
  }
  __syncthreads();

  float pq = 0.f, pk = 0.f;
  for (int t = tid; t < T_; t += 128) { pq += sl[t] * gQ1[t]; pk += sl[t] * gK1[t]; }

  red[tid] = pq; __syncthreads();
  for (int off = 64; off > 0; off >>= 1) {
    if (tid < off) red[tid] += red[tid + off];
    __syncthreads();
  }
  const float aq = red[0]; __syncthreads();
  red[tid] = pk; __syncthreads();
  for (int off = 64; off > 0; off >>= 1) {
    if (tid < off) red[tid] += red[tid + off];
    __syncthreads();
  }
  const float ak = red[0]; __syncthreads();

  if (tid < F_C) {
    qg[tid] = fmaxf(wsum[tid] * aq, 0.f);
    kg[tid] = fmaxf(wsum[tid] * ak, 0.f);
  }
  __syncthreads();

  {
    float accq = 0.f, acck = 0.f;
    for (int f = 0; f < F_C; ++f) {
      accq += qg[f] * gQ2[tid * F_C + f];
      acck += kg[f] * gK2[tid * F_C + f];
    }
    Qg[tid] = accq;
    Kg_out[(size_t)bn * 128 + tid] = acck;
  }
  __syncthreads();

  {
    float acc = 0.f;
    for (int h = 0; h < 128; ++h) acc += Qg[h] * gWa[h * 128 + tid];
    Qa_out[(size_t)bn * 128 + tid] = acc;
  }
}

// ---------------------------------------------------------------------------
// K4: per (b,n): A[m] = Qa . Kg[b,m]; softmax over m; adj2 = A * adj[n,:].
// 352 threads (11 waves), reduction padded to 512.
// ---------------------------------------------------------------------------
__global__ __launch_bounds__(352) void k_adj2(
    const float* __restrict__ Qa, const float* __restrict__ Kg,
    const float* __restrict__ adj, float* __restrict__ adj2_out) {
  const int bn = blockIdx.x, b = bn / N_, n = bn % N_, tid = threadIdx.x;
  __shared__ float qa[128];
  __shared__ float red[512];
  __shared__ float smax, ssum;

  if (tid < 128) qa[tid] = Qa[(size_t)bn * 128 + tid];
  __syncthreads();

  float val = -1e30f;
  if (tid < N_) {
    const float* kgr = Kg + ((size_t)b * N_ + tid) * 128;
    float acc = 0.f;
    for (int g = 0; g < 128; ++g) acc += qa[g] * kgr[g];
    val = acc;
  }
  red[tid] = val;
  for (int i = 352 + tid; i < 512; i += 352) red[i] = -1e30f;
  __syncthreads();
  for (int off = 256; off > 0; off >>= 1) {
    if (tid < off && tid + off < 512) red[tid] = fmaxf(red[tid], red[tid + off]);
    __syncthreads();
  }
  if (tid == 0) smax = red[0];
  __syncthreads();

  const float e = (tid < N_) ? expf(val - smax) : 0.f;
  red[tid] = e;
  for (int i = 352 + tid; i < 512; i += 352) red[i] = 0.f;
  __syncthreads();
  for (int off = 256; off > 0; off >>= 1) {
    if (tid < off && tid + off < 512) red[tid] += red[tid + off];
    __syncthreads();
  }
  if (tid == 0) ssum = red[0];
  __syncthreads();

  if (tid < N_)
    adj2_out[(size_t)bn * N_ + tid] = (e / ssum) * adj[(size_t)n * N_ + tid];
}

// ---------------------------------------------------------------------------
// K5: per (b,n): c[t] = adj2_row . s[b,:,t]; then chunked gelu + WMMA GEMM
//   new_values[t, f] = sum_{k<128} gelu(gw[k]*c[t]) * gWout[k, f]
//
// 256 threads = 8 waves.  t processed in 9 chunks of 32 rows:
//   phase a) all threads fill G[32][128] = gelu(gw*c) in LDS (gelu once per
//            (t,k); divergent erf code fully outside the WMMA region)
//   phase b) each wave computes exactly one 16x16 tile: t-tile = wave>>2,
//            f-tile = wave&3.  B-fragments (depend only on f-tile) live in
//            registers for the whole kernel; A-fragments are batch-loaded as
//            32 ds_load_b64 then one wait, then 32 chained v_wmma.
//
// V_WMMA_F32_16X16X4_F32 layout (ISA 7.12.2):
//   A 16x4 : M = lane&15, K = vgpr_j + 2*(lane>>4)
//   B 4x16 : N = lane&15, K = vgpr_j + 2*(lane>>4)   (mirror assumption)
//   D 16x16: VGPR r -> M = r + 8*(lane>>4), N = lane&15
// ---------------------------------------------------------------------------
__global__ __launch_bounds__(256) void k_gcn_out(
    const float* __restrict__ adj2, const float* __restrict__ s,
    const float* __restrict__ Wout_map, const float* __restrict__ gWg,
    const float* __restrict__ gWout, float* __restrict__ nv) {
  const int bn = blockIdx.x, b = bn / N_, tid = threadIdx.x;
  __shared__ float arow[N_];
  __shared__ float cvec[T_];
  __shared__ float gw[128];
  __shared__ __align__(16) float Gt[32 * 128];  // 16 KB chunk of G

  for (int i = tid; i < N_; i += blockDim.x)
    arow[i] = adj2[(size_t)bn * N_ + i];
  if (tid < 128) {
    // gw[h] = sum_f wsum[f] * gWg[f,h]
    float acc = 0.f;
    for (int f = 0; f < F_C; ++f) {
      float w = 0.f;
#pragma unroll
      for (int c = 0; c < FPH_; ++c) w += Wout_map[f * FPH_ + c];
      acc += w * gWg[f * 128 + tid];
    }
    gw[tid] = acc;
  }
  __syncthreads();

  // c[t] = sum_m arow[m] * s[b, m, t]   (coalesced across lanes over t)
  {
    const float* sb = s + (size_t)b * N_ * T_;
    for (int t = tid; t < T_; t += blockDim.x) {
      float acc = 0.f;
      for (int m = 0; m < N_; ++m) {
        if (m + 8 < N_) __builtin_prefetch(&sb[(size_t)(m + 8) * T_ + t], 0, 0);
        acc += arow[m] * sb[(size_t)m * T_ + t];
      }
      cvec[t] = acc;
    }
  }
  __syncthreads();

  const int wave = tid >> 5;
  const int lane = tid & 31;
  const int mrow = lane & 15;   // M (A/D) or N (B/D column)
  const int grp  = lane >> 4;   // half-wave group
  const int f0   = (wave & 3) * 16;
  const int trow = (wave >> 2) * 16;  // t-tile base inside a chunk

  // Hoist B-fragments for this wave's f-column into registers: per k-step
  // kk, lane needs gWout[(4kk+2g)*64 + f0+n] and the next k row.
  float breg[64];
  {
    const float* wcol = gWout + f0 + mrow;
#pragma unroll
    for (int kk = 0; kk < 32; ++kk) {
      const int kb = 4 * kk + 2 * grp;
      breg[2 * kk]     = wcol[(size_t)kb * F_C];
      breg[2 * kk + 1] = wcol[(size_t)(kb + 1) * F_C];
    }
  }

  for (int chunk = 0; chunk < 9; ++chunk) {
    const int tbase = chunk * 32;
    // phase a: G[r][h] = gelu(gw[h] * c[tbase + r]) for r<32, h<128
    for (int i = tid; i < 32 * 128; i += blockDim.x) {
      const int r = i >> 7, h = i & 127;
      Gt[i] = gelu_exact(gw[h] * cvec[tbase + r]);
    }
    __syncthreads();

    // phase b: one 16x16 tile per wave, A-frags batch-loaded then 32 wmma
    v2f afr[32];
#pragma unroll
    for (int kk = 0; kk < 32; ++kk) {
      afr[kk] = *reinterpret_cast<const v2f*>(
          &Gt[(trow + mrow) * 128 + 4 * kk + 2 * grp]);
    }
    v8f acc = {};
#pragma unroll
    for (int kk = 0; kk < 32; ++kk) {
      v2f bb;
      bb.x = breg[2 * kk];
      bb.y = breg[2 * kk + 1];
      acc = __builtin_amdgcn_wmma_f32_16x16x4_f32(
          /*neg_a=*/false, afr[kk], /*neg_b=*/false, bb,
          /*c_mod=*/(short)0, acc, /*reuse_a=*/false, /*reuse_b=*/false);
    }
    float* outb = nv + ((size_t)bn * T_ + tbase + trow) * F_C + f0;
#pragma unroll
    for (int r = 0; r < 8; ++r) {
      const int M = r + 8 * grp;
      outb[(size_t)M * F_C + mrow] = acc[r];
    }
    __syncthreads();  // protect Gt before next chunk overwrites it
  }
}

// ---------------------------------------------------------------------------
extern "C" void kernel_launch(void* const* d_in, const int* in_sizes, int n_in,
                              void* d_out, int out_size, void* d_ws,
                              size_t ws_size, hipStream_t stream) {
  const float* Q_in     = (const float*)d_in[0];
  const float* K_in     = (const float*)d_in[1];
  const float* V_in     = (const float*)d_in[2];
  const float* adj      = (const float*)d_in[3];
  const float* Wq       = (const float*)d_in[4];
  const float* Wk       = (const float*)d_in[5];
  const float* Wv       = (const float*)d_in[6];
  const float* Wout_map = (const float*)d_in[7];
  const float* gQ1      = (const float*)d_in[8];
  const float* gQ2      = (const float*)d_in[9];
  const float* gK1      = (const float*)d_in[10];
  const float* gK2      = (const float*)d_in[11];
  const float* gWa      = (const float*)d_in[12];
  const float* gWg      = (const float*)d_in[13];
  const float* gWout    = (const float*)d_in[14];

  // Output layout: new_values | delay | tmp_corr | adj2 (flat, return order)
  const size_t NV_SZ  = (size_t)BN_ * T_ * F_C;      // 23,961,600
  const size_t DL_SZ  = (size_t)BN_ * H_ * TOPK_;    // 52,000
  float* nv        = (float*)d_out;
  float* delay_out = nv + NV_SZ;
  float* tc_out    = delay_out + DL_SZ;
  float* adj2_out  = tc_out + DL_SZ;

  // Workspace layout
  const size_t VS_SZ = (size_t)ROWS_ * FPH_;  // 2,995,200
  const size_t S_SZ  = (size_t)ROWS_;         //   374,400
  const size_t QA_SZ = (size_t)BN_ * 128;     //   166,400
  float* ws  = (float*)d_ws;
  float* vsm = ws;
  float* sgl = vsm + VS_SZ;
  float* Qa  = sgl + S_SZ;
  float* Kgw = Qa + QA_SZ;

  k_vsmall<<<(ROWS_ + 255) / 256, 256, 0, stream>>>(V_in, Wv, vsm);
  k_corr<<<BN_, 288, 0, stream>>>(Q_in, K_in, Wq, Wk, vsm, sgl, delay_out,
                                  tc_out);
  k_gattn<<<BN_, 128, 0, stream>>>(sgl, Wout_map, gQ1, gQ2, gK1, gK2, gWa, Qa,
                                   Kgw);
  k_adj2<<<BN_, 352, 0, stream>>>(Qa, Kgw, adj, adj2_out);
  k_gcn_out<<<BN_, 256, 0, stream>>>(adj2_out, sgl, Wout_map, gWg, gWout, nv);
}